// GeodesicConvImg_50019189129843
// MI455X (gfx1250) — compile-verified
//
#include <hip/hip_runtime.h>
#include <math.h>

typedef float v2f __attribute__((ext_vector_type(2)));
typedef float v8f __attribute__((ext_vector_type(8)));

namespace {
constexpr int H = 128, W = 128, ND = 8, C = 16;
constexpr int ESTRIDE = 18;                       // dwords per (r,e) row: 16 ch + 2 pad (even -> 8B-aligned b64)
constexpr int CENTER_OFF = 4 * 8 * ESTRIDE;       // 576 dwords (ring region)
constexpr int PSTRIDE = CENTER_OFF + 8 * ESTRIDE + 4; // 724 dwords per pixel (pad for bank spread)
constexpr int WAVES = 4;                          // waves per block (128 threads)
constexpr int KTOT = 528;                         // 512 ring K + 16 center K
constexpr int NQ = KTOT / 2;                      // 264 interleaved row-pairs of B
}

__global__ __launch_bounds__(128, 1)
void geoconv_wmma(const float* __restrict__ img,   // (B,H,W,D,C) f32
                  const float* __restrict__ kw,    // (4,16,16,16) f32
                  const float* __restrict__ ck,    // (16,16) f32
                  const float* __restrict__ bias,  // (16,) f32
                  float* __restrict__ out)         // (B,H,W,16) f32
{
  __shared__ __align__(16) float sW[NQ * 32];             // 33792 B: W[k][f], pairs interleaved
  __shared__ __align__(16) float sP[WAVES * 2 * PSTRIDE]; // 23168 B: patches per pixel

  const int tid = threadIdx.x;

  // ---- stage 528x16 weight matrix, row-pair interleaved: sW[(k>>1)*32 + f*2 + (k&1)]
  for (int k = tid; k < KTOT; k += 128) {
    const float* src;
    if (k < 512) {
      const int r = k >> 7, ep = (k >> 4) & 7, c = k & 15;
      src = kw + ((r * 16 + 2 * ep) * 16 + c) * 16;   // kernel_weights[r, 2*e', c, :]
    } else {
      src = ck + (k - 512) * 16;                      // center_kernel[c, :]
    }
    float* dst = sW + (k >> 1) * 32 + (k & 1);
    #pragma unroll
    for (int f = 0; f < 16; ++f) dst[2 * f] = src[f];
  }

  const int wave = tid >> 5;
  const int lane = tid & 31;
  const int ch   = lane & 15;     // channel this lane samples
  const int half = lane >> 4;     // which of the wave's 2 pixels
  const int pixPair = blockIdx.x * WAVES + wave;
  const int pix = pixPair * 2 + half;
  const int x = pix & (W - 1);
  const int y = (pix >> 7) & (H - 1);
  const int b = pix >> 14;
  float* pb = sP + (wave * 2 + half) * PSTRIDE;

  const float TWO_PI_F = 6.2831854820251464844f;

  // ---- sample rings (r=1..3) + center (== ring r=0) into LDS
  for (int t = 0; t < 32; ++t) {
    int e, rr; float dx, dy;
    if (t < 24) {
      rr = 1 + (t >> 3); e = t & 7;
      const float ang = TWO_PI_F * (float)e * 0.125f;
      dx = (float)rr * cosf(ang);
      dy = (float)rr * sinf(ang);
    } else {
      rr = 0; e = t - 24; dx = 0.f; dy = 0.f;
    }
    const float ex = 0.5f + (float)x + dx;
    const float ey = 0.5f + (float)y + dy;
    const float fx = floorf(ex), fy = floorf(ey);
    const float wx = ex - fx,   wy = ey - fy;
    const int x0 = (int)fx, y0 = (int)fy;
    const int x0i = min(max(x0, 0), W - 1), x1i = min(max(x0 + 1, 0), W - 1);
    const int y0i = min(max(y0, 0), H - 1), y1i = min(max(y0 + 1, 0), H - 1);
    auto at = [&](int yy, int xx) -> float {
      return img[(((b * H + yy) * W + xx) * ND + e) * C + ch];
    };
    const float v = at(y0i, x0i) * (1.f - wy) * (1.f - wx)
                  + at(y1i, x0i) * wy * (1.f - wx)
                  + at(y0i, x1i) * (1.f - wy) * wx
                  + at(y1i, x1i) * wy * wx;
    if (t < 24) {
      pb[(rr * 8 + e) * ESTRIDE + ch] = v;
    } else {
      pb[CENTER_OFF + e * ESTRIDE + ch] = v;  // center row (K 512..527)
      pb[e * ESTRIDE + ch] = v;               // ring r=0 is identical (dx=dy=0)
    }
  }
  __syncthreads();

  // ---- GEMM via V_WMMA_F32_16X16X4_F32: 16 rows = (2 pixels x 8 dirs), K=528, N=16
  const int M  = lane & 15;       // tile row
  const int d  = M & 7;           // output direction (rotation amount)
  const int pA = M >> 3;          // which pixel this row belongs to
  const int hi = lane >> 4;       // lanes 16-31 carry K+2,K+3 halves
  const int f  = lane & 15;       // N index for B/D
  const float* aPix = sP + (wave * 2 + pA) * PSTRIDE;
  const float* bPtr = sW + hi * 32 + f * 2;

  v8f acc = {0.f, 0.f, 0.f, 0.f, 0.f, 0.f, 0.f, 0.f};
  int s = 0;  // K-step index (K = 4*s)
  for (int r = 0; r < 4; ++r) {
    for (int ep = 0; ep < 8; ++ep) {
      const int e = (ep + d) & 7;                         // rotated patch direction
      const float* ab = aPix + (r * 8 + e) * ESTRIDE + hi * 2;
      #pragma unroll
      for (int cc = 0; cc < 4; ++cc) {
        v2f a  = *(const v2f*)(ab + cc * 4);              // A[M][4s+2hi .. +1]  (ds_load_b64)
        v2f bv = *(const v2f*)(bPtr + s * 64);            // B[4s+2hi .. +1][f]  (ds_load_b64)
        acc = __builtin_amdgcn_wmma_f32_16x16x4_f32(
            false, a, false, bv, (short)0, acc, false, false);
        ++s;
      }
    }
  }
  { // center K rows 512..527: A row = center[pA, d, c]
    const float* ab = aPix + CENTER_OFF + d * ESTRIDE + hi * 2;
    #pragma unroll
    for (int cc = 0; cc < 4; ++cc) {
      v2f a  = *(const v2f*)(ab + cc * 4);
      v2f bv = *(const v2f*)(bPtr + s * 64);
      acc = __builtin_amdgcn_wmma_f32_16x16x4_f32(
          false, a, false, bv, (short)0, acc, false, false);
      ++s;
    }
  }

  // ---- bias + relu + max over d (D layout: vgpr v -> M=v | M=v+8)
  const float bb = bias[f];
  float m = 0.f;                      // relu floor doubles as max identity
  #pragma unroll
  for (int v = 0; v < 8; ++v) m = fmaxf(m, acc[v] + bb);

  const int opix = pixPair * 2 + hi;  // lanes<16 -> pixel0, lanes>=16 -> pixel1
  out[opix * 16 + f] = m;             // 32 lanes -> 128 contiguous bytes
}

extern "C" void kernel_launch(void* const* d_in, const int* in_sizes, int n_in,
                              void* d_out, int out_size, void* d_ws, size_t ws_size,
                              hipStream_t stream) {
  const float* img  = (const float*)d_in[0];  // (4,128,128,8,16)
  const float* kw   = (const float*)d_in[1];  // (4,16,16,16)
  const float* ck   = (const float*)d_in[2];  // (16,16)
  const float* bias = (const float*)d_in[3];  // (16,)
  float* out = (float*)d_out;                 // (4,128,128,16)

  const int pixPairs = 4 * 128 * 128 / 2;     // 32768 (2 pixels per wave)
  const int blocks = pixPairs / WAVES;        // 8192 blocks of 4 waves
  geoconv_wmma<<<blocks, 128, 0, stream>>>(img, kw, ck, bias, out);
}